// Decoder_20066087207436
// MI455X (gfx1250) — compile-verified
//
#include <hip/hip_runtime.h>
#include <math.h>

typedef __attribute__((ext_vector_type(2))) float v2f;
typedef __attribute__((ext_vector_type(8))) float v8f;

#define BATCH   64
#define SEQ     512
#define NUM_VEC 512
#define NUM_HID 1024
#define NUM_OUT 32000

#define RNN_WGS     (NUM_HID / 16)   // 64 workgroups, each owns 16 hidden units
#define RNN_THREADS 128              // 4 waves -> 4 M-tiles of 16 (BATCH=64)

#define WMMA_F32(A, B, C) \
    __builtin_amdgcn_wmma_f32_16x16x4_f32(false, (A), false, (B), (short)0, (C), false, false)

// ---------------------------------------------------------------------------
// Init: zero h0 and the grid-barrier counter (runs every call -> deterministic)
// ---------------------------------------------------------------------------
__global__ void init_kernel(float* __restrict__ hA, unsigned int* __restrict__ ctr) {
    int i = blockIdx.x * blockDim.x + threadIdx.x;
    if (i < BATCH * NUM_HID) hA[i] = 0.0f;
    if (i == 0) *ctr = 0u;
}

// ---------------------------------------------------------------------------
// Persistent recurrent kernel: 512 steps of h = tanh(x_t W_ih^T + h W_hh^T + b)
// Weights stationary in LDS. fp32 WMMA 16x16x4, grouped so that 8 fragment
// loads (distinct registers) are in flight per 8 WMMAs, with 4 independent
// accumulator chains (dep distance 4).
// ---------------------------------------------------------------------------
__global__ void rnn_persistent(const float* __restrict__ X,
                               const float* __restrict__ W_ih,
                               const float* __restrict__ b_ih,
                               const float* __restrict__ W_hh,
                               const float* __restrict__ b_hh,
                               float* __restrict__ hA,
                               float* __restrict__ hB,
                               unsigned int* __restrict__ barrier_ctr)
{
    extern __shared__ float lds[];
    float* sWi   = lds;                    // [NUM_VEC][16]  (K-major, transposed)
    float* sWh   = sWi + NUM_VEC * 16;     // [NUM_HID][16]
    float* sBias = sWh + NUM_HID * 16;     // [16]

    const int wg      = blockIdx.x;        // 0..63
    const int tid     = threadIdx.x;       // 0..127
    const int colBase = wg * 16;           // hidden-unit slice owned by this WG

    // Preload transposed weight slices into LDS (once; reused 512 steps).
    for (int i = tid; i < NUM_VEC * 16; i += blockDim.x) {
        int k = i >> 4, jj = i & 15;
        sWi[i] = W_ih[(colBase + jj) * NUM_VEC + k];
    }
    for (int i = tid; i < NUM_HID * 16; i += blockDim.x) {
        int k = i >> 4, jj = i & 15;
        sWh[i] = W_hh[(colBase + jj) * NUM_HID + k];
    }
    if (tid < 16) sBias[tid] = b_ih[colBase + tid] + b_hh[colBase + tid];
    __syncthreads();

    const int wave  = tid >> 5;            // 0..3 -> batch-row tile (M)
    const int lane  = tid & 31;
    const int n     = lane & 15;           // N (and A-matrix M-row) index
    const int khalf = (lane >> 4) * 2;     // K sub-offset: 0 or 2 per ISA layout

    const float* hcur = hA;
    float*       hnxt = hB;

    const float* xrowBase = X + (size_t)(wave * 16 + n) * (SEQ * NUM_VEC);

    for (int t = 0; t < SEQ; ++t) {
        v8f acc0 = {}, acc1 = {}, acc2 = {}, acc3 = {};

        // ---- x_t contribution: A = x_t (64 x 512), B = W_ih slice (512 x 16)
        const float* xrow = xrowBase + (size_t)t * NUM_VEC;
        for (int k0 = 0; k0 < NUM_VEC; k0 += 32) {
            v2f a[8], b[8];
#pragma unroll
            for (int u = 0; u < 8; ++u) {
                const int k = k0 + 4 * u + khalf;
                a[u].x = xrow[k];
                a[u].y = xrow[k + 1];
                b[u].x = sWi[k * 16 + n];
                b[u].y = sWi[(k + 1) * 16 + n];
            }
            acc0 = WMMA_F32(a[0], b[0], acc0);
            acc1 = WMMA_F32(a[1], b[1], acc1);
            acc2 = WMMA_F32(a[2], b[2], acc2);
            acc3 = WMMA_F32(a[3], b[3], acc3);
            acc0 = WMMA_F32(a[4], b[4], acc0);
            acc1 = WMMA_F32(a[5], b[5], acc1);
            acc2 = WMMA_F32(a[6], b[6], acc2);
            acc3 = WMMA_F32(a[7], b[7], acc3);
        }
        // ---- h contribution: A = h (64 x 1024), B = W_hh slice (1024 x 16)
        const float* hrow = hcur + (size_t)(wave * 16 + n) * NUM_HID;
        for (int k0 = 0; k0 < NUM_HID; k0 += 32) {
            v2f a[8], b[8];
#pragma unroll
            for (int u = 0; u < 8; ++u) {
                const int k = k0 + 4 * u + khalf;
                a[u].x = hrow[k];
                a[u].y = hrow[k + 1];
                b[u].x = sWh[k * 16 + n];
                b[u].y = sWh[(k + 1) * 16 + n];
            }
            acc0 = WMMA_F32(a[0], b[0], acc0);
            acc1 = WMMA_F32(a[1], b[1], acc1);
            acc2 = WMMA_F32(a[2], b[2], acc2);
            acc3 = WMMA_F32(a[3], b[3], acc3);
            acc0 = WMMA_F32(a[4], b[4], acc0);
            acc1 = WMMA_F32(a[5], b[5], acc1);
            acc2 = WMMA_F32(a[6], b[6], acc2);
            acc3 = WMMA_F32(a[7], b[7], acc3);
        }

        // ---- combine chains, bias + tanh, scatter C (row r / r+8 per lane half)
        const float bias = sBias[n];
        const int   rofs = (lane >> 4) * 8;
#pragma unroll
        for (int r = 0; r < 8; ++r) {
            int brow = wave * 16 + rofs + r;
            float v = (acc0[r] + acc1[r]) + (acc2[r] + acc3[r]) + bias;
            hnxt[(size_t)brow * NUM_HID + colBase + n] = tanhf(v);
        }

        // ---- grid-wide barrier (monotonic counter, reset by init_kernel)
        __threadfence();
        __syncthreads();
        if (tid == 0) {
            atomicAdd(barrier_ctr, 1u);
            const unsigned target = (unsigned)gridDim.x * (unsigned)(t + 1);
            while (atomicAdd(barrier_ctr, 0u) < target) {
                __builtin_amdgcn_s_sleep(1);
            }
        }
        __syncthreads();
        __threadfence();

        float* tmp = (float*)hcur; hcur = hnxt; hnxt = tmp;
    }
    // SEQ is even -> final hidden state ends up in hA.
}

// ---------------------------------------------------------------------------
// logits = h_final @ W_out^T + b_out  (HBM-bound: 131 MB of W_out read once)
// One wave per 16 output cols; computes a 64x16 tile with fp32 WMMA.
// Fragment loads grouped ahead of the 8-WMMA burst; 4 independent accumulators.
// ---------------------------------------------------------------------------
__global__ void logits_kernel(const float* __restrict__ hfin,
                              const float* __restrict__ W_out,
                              const float* __restrict__ b_out,
                              float* __restrict__ out)
{
    const int nBase = blockIdx.x * 16;     // 2000 blocks
    const int lane  = threadIdx.x;         // 32 threads = 1 wave
    const int n     = lane & 15;
    const int khalf = (lane >> 4) * 2;

    v8f acc0 = {}, acc1 = {}, acc2 = {}, acc3 = {};
    const float* wrow = W_out + (size_t)(nBase + n) * NUM_HID;

    for (int k0 = 0; k0 < NUM_HID; k0 += 8) {
        v2f b0, b1, a[8];
        b0.x = wrow[k0 + khalf];     b0.y = wrow[k0 + khalf + 1];
        b1.x = wrow[k0 + 4 + khalf]; b1.y = wrow[k0 + 4 + khalf + 1];
#pragma unroll
        for (int mt = 0; mt < 4; ++mt) {
            const float* hm = hfin + (size_t)(mt * 16 + n) * NUM_HID;
            a[mt].x     = hm[k0 + khalf];     a[mt].y     = hm[k0 + khalf + 1];
            a[4 + mt].x = hm[k0 + 4 + khalf]; a[4 + mt].y = hm[k0 + 4 + khalf + 1];
        }
        acc0 = WMMA_F32(a[0], b0, acc0);
        acc1 = WMMA_F32(a[1], b0, acc1);
        acc2 = WMMA_F32(a[2], b0, acc2);
        acc3 = WMMA_F32(a[3], b0, acc3);
        acc0 = WMMA_F32(a[4], b1, acc0);
        acc1 = WMMA_F32(a[5], b1, acc1);
        acc2 = WMMA_F32(a[6], b1, acc2);
        acc3 = WMMA_F32(a[7], b1, acc3);
    }

    const float bias = b_out[nBase + n];
    const int   rofs = (lane >> 4) * 8;
#pragma unroll
    for (int r = 0; r < 8; ++r) {
        out[(size_t)( 0 + rofs + r) * NUM_OUT + nBase + n] = acc0[r] + bias;
        out[(size_t)(16 + rofs + r) * NUM_OUT + nBase + n] = acc1[r] + bias;
        out[(size_t)(32 + rofs + r) * NUM_OUT + nBase + n] = acc2[r] + bias;
        out[(size_t)(48 + rofs + r) * NUM_OUT + nBase + n] = acc3[r] + bias;
    }
}

// ---------------------------------------------------------------------------
// In-place row softmax over 32000 elements, one block per batch row.
// ---------------------------------------------------------------------------
__global__ void softmax_kernel(float* __restrict__ out)
{
    __shared__ float red[256];
    const int tid = threadIdx.x;
    float* p = out + (size_t)blockIdx.x * NUM_OUT;

    float m = -INFINITY;
    for (int i = tid; i < NUM_OUT; i += 256) m = fmaxf(m, p[i]);
    red[tid] = m; __syncthreads();
    for (int s = 128; s > 0; s >>= 1) {
        if (tid < s) red[tid] = fmaxf(red[tid], red[tid + s]);
        __syncthreads();
    }
    m = red[0]; __syncthreads();

    float sum = 0.0f;
    for (int i = tid; i < NUM_OUT; i += 256) {
        float e = expf(p[i] - m);
        p[i] = e;
        sum += e;
    }
    red[tid] = sum; __syncthreads();
    for (int s = 128; s > 0; s >>= 1) {
        if (tid < s) red[tid] += red[tid + s];
        __syncthreads();
    }
    const float inv = 1.0f / red[0];
    __syncthreads();
    for (int i = tid; i < NUM_OUT; i += 256) p[i] *= inv;
}

// ---------------------------------------------------------------------------
extern "C" void kernel_launch(void* const* d_in, const int* in_sizes, int n_in,
                              void* d_out, int out_size, void* d_ws, size_t ws_size,
                              hipStream_t stream)
{
    (void)in_sizes; (void)n_in; (void)out_size; (void)ws_size;

    const float* X     = (const float*)d_in[0];
    const float* W_ih  = (const float*)d_in[1];
    const float* b_ih  = (const float*)d_in[2];
    const float* W_hh  = (const float*)d_in[3];
    const float* b_hh  = (const float*)d_in[4];
    const float* W_out = (const float*)d_in[5];
    const float* b_out = (const float*)d_in[6];
    float*       out   = (float*)d_out;

    // Workspace layout: hA | hB | barrier counter   (~512 KB + 4 B)
    float* hA = (float*)d_ws;
    float* hB = hA + BATCH * NUM_HID;
    unsigned int* ctr = (unsigned int*)(hB + BATCH * NUM_HID);

    init_kernel<<<(BATCH * NUM_HID + 255) / 256, 256, 0, stream>>>(hA, ctr);

    const size_t shmem = (size_t)(NUM_VEC * 16 + NUM_HID * 16 + 16) * sizeof(float); // ~96 KB
    hipFuncSetAttribute((const void*)rnn_persistent,
                        hipFuncAttributeMaxDynamicSharedMemorySize, (int)shmem);
    rnn_persistent<<<RNN_WGS, RNN_THREADS, shmem, stream>>>(
        X, W_ih, b_ih, W_hh, b_hh, hA, hB, ctr);

    // SEQ even -> final h in hA
    logits_kernel<<<NUM_OUT / 16, 32, 0, stream>>>(hA, W_out, b_out, out);
    softmax_kernel<<<BATCH, 256, 0, stream>>>(out);
}